// DEMA_13511967113406
// MI455X (gfx1250) — compile-verified
//
#include <hip/hip_runtime.h>

typedef float v2f __attribute__((ext_vector_type(2)));
typedef float v8f __attribute__((ext_vector_type(8)));

#define T_DIM 4096
#define C_DIM 512
#define B_DIM 32
#define CHUNK 16
#define NCHUNK 255   // 255 full chunks of 16 steps; 15-step scalar tail

__global__ __launch_bounds__(256) void dema_wmma_kernel(
    const float* __restrict__ x, const float* __restrict__ alpha,
    const float* __restrict__ beta, float* __restrict__ out)
{
    __shared__ float sW[16];   // w_m = (M^m c)[0], m=0..15
    __shared__ float sU[16];   // u_m = (M^m c)[1]
    __shared__ float sP[16];   // p_k = (M^k)[0][0], k=1..16 at idx k-1
    __shared__ float sQ[16];   // q_k = (M^k)[0][1]
    __shared__ float sR[2];    // row1 of M^16

    const float a  = alpha[0];
    const float bt = beta[0];

    if (threadIdx.x == 0) {
        const float m00 = 1.0f - a, m01 = 1.0f - a;
        const float m10 = -a * bt,  m11 = 1.0f - a * bt;
        float cs = a, cb = a * bt;                 // M^0 c
        for (int m = 0; m < 16; ++m) {
            sW[m] = cs; sU[m] = cb;
            float ns = m00 * cs + m01 * cb;
            float nb = m10 * cs + m11 * cb;
            cs = ns; cb = nb;
        }
        float p00 = m00, p01 = m01, p10 = m10, p11 = m11;  // M^1
        for (int k = 1; k <= 16; ++k) {
            sP[k - 1] = p00; sQ[k - 1] = p01;
            if (k == 16) { sR[0] = p10; sR[1] = p11; break; }
            float n00 = m00 * p00 + m01 * p10;
            float n01 = m00 * p01 + m01 * p11;
            float n10 = m10 * p00 + m11 * p10;
            float n11 = m10 * p01 + m11 * p11;
            p00 = n00; p01 = n01; p10 = n10; p11 = n11;
        }
    }
    __syncthreads();

    const int lane   = threadIdx.x & 31;
    const int lane16 = lane & 15;
    const int hi     = lane >> 4;                      // lane half: 0 or 1
    const int wid    = blockIdx.x * (blockDim.x >> 5) + (threadIdx.x >> 5);
    const int bb     = wid >> 5;                       // batch 0..31
    const int cg     = wid & 31;                       // channel group 0..31
    const int c0     = cg * 16;
    const int ch     = c0 + lane16;                    // this lane's channel

    const float* xb = x   + (size_t)bb * T_DIM * C_DIM;
    float*       ob = out + (size_t)bb * T_DIM * C_DIM;

    // ---- per-lane invariant A fragments: A[M][K] = (K<=M) ? w[M-K] : 0
    v2f afrag[4];
#pragma unroll
    for (int kb = 0; kb < 4; ++kb) {
        int K0 = kb * 4 + hi * 2;
        int M  = lane16;
        afrag[kb].x = (K0     <= M) ? sW[M - K0]     : 0.0f;
        afrag[kb].y = (K0 + 1 <= M) ? sW[M - K0 - 1] : 0.0f;
    }
    // boundary coefficients for the 8 D-rows this lane owns
    float pReg[8], qReg[8];
#pragma unroll
    for (int v = 0; v < 8; ++v) {
        int r = v + hi * 8;
        pReg[v] = sP[r]; qReg[v] = sQ[r];
    }
    // b-state dot coefficients for the 8 x-rows this lane holds
    float uReg[8];
#pragma unroll
    for (int kb = 0; kb < 4; ++kb) {
        int K0 = kb * 4 + hi * 2;
        uReg[kb * 2]     = sU[15 - K0];
        uReg[kb * 2 + 1] = sU[15 - K0 - 1];
    }
    const float r0 = sR[0], r1 = sR[1];

    // ---- initial state: s0 = x[b,0,c], b0 = x[b,1,c] - s0; out[b,0,c] = s0
    float s_st = xb[ch];
    float b_st = xb[C_DIM + ch] - s_st;
    if (hi == 0) __builtin_nontemporal_store(s_st, ob + ch);

    for (int chn = 0; chn < NCHUNK; ++chn) {
        const int t0 = chn * CHUNK;
        const float* xt = xb + (size_t)(t0 + 1) * C_DIM + c0 + lane16;
        float*       ot = ob + (size_t)(t0 + 1) * C_DIM + c0 + lane16;

        // hint: pull next chunk's tile toward the caches while we compute
        __builtin_prefetch(xt + (size_t)CHUNK * C_DIM + (size_t)(hi * 8) * C_DIM, 0, 1);

        // ---- B fragments: B[K][N] = x[b, t0+1+K, c0+N]
        v2f bfrag[4];
#pragma unroll
        for (int kb = 0; kb < 4; ++kb) {
            int K0 = kb * 4 + hi * 2;
            bfrag[kb].x = xt[(size_t)K0 * C_DIM];
            bfrag[kb].y = xt[(size_t)(K0 + 1) * C_DIM];
        }

        // ---- accumulator pre-loaded with boundary terms p_k*s + q_k*b
        v8f acc;
#pragma unroll
        for (int v = 0; v < 8; ++v) acc[v] = pReg[v] * s_st + qReg[v] * b_st;

        // ---- S += W * X : four chained fp32 WMMAs (K = 4 each)
        acc = __builtin_amdgcn_wmma_f32_16x16x4_f32(false, afrag[0], false, bfrag[0], (short)0, acc, false, false);
        acc = __builtin_amdgcn_wmma_f32_16x16x4_f32(false, afrag[1], false, bfrag[1], (short)0, acc, false, false);
        acc = __builtin_amdgcn_wmma_f32_16x16x4_f32(false, afrag[2], false, bfrag[2], (short)0, acc, false, false);
        acc = __builtin_amdgcn_wmma_f32_16x16x4_f32(false, afrag[3], false, bfrag[3], (short)0, acc, false, false);

        // ---- store 16 output rows (streaming / non-temporal)
#pragma unroll
        for (int v = 0; v < 8; ++v) {
            int r = v + hi * 8;
            __builtin_nontemporal_store(acc[v], ot + (size_t)r * C_DIM);
        }

        // ---- state update
        float bp = uReg[0] * bfrag[0].x + uReg[1] * bfrag[0].y
                 + uReg[2] * bfrag[1].x + uReg[3] * bfrag[1].y
                 + uReg[4] * bfrag[2].x + uReg[5] * bfrag[2].y
                 + uReg[6] * bfrag[3].x + uReg[7] * bfrag[3].y;
        bp += __shfl_xor(bp, 16, 32);                       // cross-half reduce
        float s_new = __shfl(acc[7], lane16 + 16, 32);      // D row 15 lives in hi half, v=7
        b_st = r0 * s_st + r1 * b_st + bp;
        s_st = s_new;
    }

    // ---- scalar tail: outputs t = 4081..4095 (15 steps), low half-wave only
    if (hi == 0) {
        float s = s_st, b = b_st;
        const float one_a  = 1.0f - a;
        const float one_bt = 1.0f - bt;
        for (int t = NCHUNK * CHUNK + 1; t < T_DIM; ++t) {
            float xv = xb[(size_t)t * C_DIM + ch];
            float sp = s;
            s = a * xv + one_a * (sp + b);
            b = bt * (s - sp) + one_bt * b;
            __builtin_nontemporal_store(s, ob + (size_t)t * C_DIM + ch);
        }
    }
}

extern "C" void kernel_launch(void* const* d_in, const int* in_sizes, int n_in,
                              void* d_out, int out_size, void* d_ws, size_t ws_size,
                              hipStream_t stream) {
    const float* x     = (const float*)d_in[0];
    const float* alpha = (const float*)d_in[1];
    const float* beta  = (const float*)d_in[2];
    float*       out   = (float*)d_out;

    // 32 batches * 32 channel-groups = 1024 wave-tasks; 8 waves (256 thr) per block
    dim3 grid(128), block(256);
    hipLaunchKernelGGL(dema_wmma_kernel, grid, block, 0, stream, x, alpha, beta, out);
}